// SoftVQLayer_33457795235999
// MI455X (gfx1250) — compile-verified
//
#include <hip/hip_runtime.h>
#include <hip/hip_bf16.h>
#include <stdint.h>

typedef __attribute__((ext_vector_type(2))) float v2f;
typedef __attribute__((ext_vector_type(8))) float v8f;

#define DIMS 128
#define NCODE 8192
#define NROW 16384
#define WAVES_PER_BLK 4
#define CHUNK 16
#define NCHUNK (NCODE / CHUNK)
#define EPSF 1e-8f

__device__ __forceinline__ float shflx16(float v) { return __shfl_xor(v, 16, 32); }

// lanes0-15 take a (their own value), lanes16-31 take b's lower-half value
__device__ __forceinline__ float comb_lo(float a, float b, int half) {
    float t = __shfl_xor(b, 16, 32);
    return half ? t : a;
}
// lanes0-15 take a's upper-half value, lanes16-31 keep b (their own value)
__device__ __forceinline__ float comb_hi(float a, float b, int half) {
    float t = __shfl_xor(a, 16, 32);
    return half ? b : t;
}

// CDNA5 async global->LDS copy (GVS mode: saddr base + per-lane 32-bit byte offset).
// VDST = LDS byte address, tracked by ASYNCcnt, no VGPR data transit.
__device__ __forceinline__ void async_copy_b128(uint32_t lds_addr,
                                                const float* gbase,
                                                int32_t voff_bytes) {
    asm volatile("global_load_async_to_lds_b128 %0, %1, %2"
                 :
                 : "v"(lds_addr), "v"(voff_bytes), "s"(gbase)
                 : "memory");
}

__device__ __forceinline__ void wait_async0() {
    asm volatile("s_wait_asynccnt 0x0" ::: "memory");
}

// flat shared-aperture pointer -> LDS byte offset (low 32 bits of the address)
__device__ __forceinline__ uint32_t lds_off(const void* p) {
    return (uint32_t)(uintptr_t)p;
}

// stage one 16x128 f32 codebook chunk (8KB) into LDS: 512 float4, 128 threads -> 4 b128 each
__device__ __forceinline__ void stage_chunk(const float* cb, int v0, float* dst, int tid) {
    const float* gsrc = cb + (size_t)v0 * DIMS;
    const uint32_t l0 = lds_off(dst);
    #pragma unroll
    for (int i = 0; i < 4; ++i) {
        const int f4 = tid + 128 * i;
        async_copy_b128(l0 + f4 * 16, gsrc, f4 * 16);
    }
}

__global__ __launch_bounds__(128) void softvq_kernel(
    const float* __restrict__ x, const float* __restrict__ cb,
    float* __restrict__ out_embed, float* __restrict__ out_idx)
{
    __shared__ float c2s[NCODE];               // 32 KB: ||c||^2 per code
    __shared__ float cstage[2][CHUNK * DIMS];  // 2 x 8 KB codebook chunk double-buffer
    const int tid = threadIdx.x;

    // prologue prefetch of chunk 0 (overlaps the c2 precompute below)
    stage_chunk(cb, 0, cstage[0], tid);

    // ---- block-wide precompute of codebook norms into LDS ----
    for (int v = tid; v < NCODE; v += blockDim.x) {
        const float4* cr = (const float4*)(cb + (size_t)v * DIMS);
        float s = 0.f;
        #pragma unroll 8
        for (int t = 0; t < DIMS / 4; ++t) {
            float4 q = cr[t];
            s += q.x * q.x + q.y * q.y + q.z * q.z + q.w * q.w;
        }
        c2s[v] = s;
    }
    __syncthreads();

    const int lane = tid & 31;
    const int wave = tid >> 5;
    const int half = lane >> 4;
    const int lm   = lane & 15;
    const int m0   = (blockIdx.x * WAVES_PER_BLK + wave) * 16;

    // ---- load 16 x-rows into WMMA B-operand layout (resident all kernel) ----
    // pair j: {x[lm][4j+2*half], x[lm][4j+2*half+1]}
    v2f xb[32];
    const float* xrow = x + (size_t)(m0 + lm) * DIMS;
    #pragma unroll
    for (int j = 0; j < 32; ++j) {
        float2 t = *(const float2*)(xrow + 4 * j + 2 * half);
        xb[j].x = t.x; xb[j].y = t.y;
    }
    // ||x||^2 per row: lane-partial (64 of 128 cols) + cross-half combine
    float x2p = 0.f;
    #pragma unroll
    for (int j = 0; j < 32; ++j) x2p += xb[j].x * xb[j].x + xb[j].y * xb[j].y;
    const float x2 = x2p + shflx16(x2p);

    // embed^T accumulators: 8 tiles of 16 dims x 16 rows
    v8f acc[8];
    #pragma unroll
    for (int t = 0; t < 8; ++t)
        acc[t] = (v8f){0.f, 0.f, 0.f, 0.f, 0.f, 0.f, 0.f, 0.f};

    float wsum  = 0.f;
    float bestd = 3.4e38f;
    int   besti = 0;

    for (int i = 0; i < NCHUNK; ++i) {
        const int v0  = i * CHUNK;
        const int cur = i & 1;

        wait_async0();       // current buffer's async copies done (own wave's)
        __syncthreads();     // ... and every other wave's too

        if (i + 1 < NCHUNK)  // prefetch next chunk into the other buffer
            stage_chunk(cb, v0 + CHUNK, cstage[cur ^ 1], tid);

        const float* __restrict__ cs = cstage[cur];

        // ---- GEMM1: S(codes x rows) = C_chunk(16xD) * X^T, A operand from LDS ----
        // Two register-staged batches of 16 operand pairs keep 8 ds_load_2addr_b64
        // in flight per batch instead of one blocking dscnt-0 per WMMA pair; two
        // accumulators break the strict D->C serial chain.
        v8f s0 = (v8f){0.f, 0.f, 0.f, 0.f, 0.f, 0.f, 0.f, 0.f};
        v8f s1 = (v8f){0.f, 0.f, 0.f, 0.f, 0.f, 0.f, 0.f, 0.f};
        const float* crow = cs + lm * DIMS;
        #pragma unroll
        for (int jj = 0; jj < 2; ++jj) {
            v2f ca[16];
            #pragma unroll
            for (int j = 0; j < 16; ++j) {
                float2 t = *(const float2*)(crow + 4 * (16 * jj + j) + 2 * half);
                ca[j].x = t.x; ca[j].y = t.y;
            }
            #pragma unroll
            for (int j = 0; j < 16; j += 2) {
                s0 = __builtin_amdgcn_wmma_f32_16x16x4_f32(
                         false, ca[j],     false, xb[16 * jj + j],     (short)0, s0, false, false);
                s1 = __builtin_amdgcn_wmma_f32_16x16x4_f32(
                         false, ca[j + 1], false, xb[16 * jj + j + 1], (short)0, s1, false, false);
            }
        }

        // ---- distances -> weights (softmax(log d) with T=1 == d/sum d) ----
        // D layout: lane holds row lm; VGPR r holds code v0 + r + 8*half
        float wr[8];
        #pragma unroll
        for (int r = 0; r < 8; ++r) {
            const int vv = v0 + r + 8 * half;
            float d2 = x2 + c2s[vv] - 2.f * (s0[r] + s1[r]);
            d2 = fmaxf(d2, 0.f);
            const float w = __builtin_sqrtf(d2) + EPSF;
            wr[r] = w;
            wsum += w;
            if (d2 < bestd) { bestd = d2; besti = vv; }
        }

        // ---- rebuild W as B operand (K-groups of 4 codes x 16 rows) ----
        v2f wb[4];
        wb[0].x = comb_lo(wr[0], wr[2], half); wb[0].y = comb_lo(wr[1], wr[3], half);
        wb[1].x = comb_lo(wr[4], wr[6], half); wb[1].y = comb_lo(wr[5], wr[7], half);
        wb[2].x = comb_hi(wr[0], wr[2], half); wb[2].y = comb_hi(wr[1], wr[3], half);
        wb[3].x = comb_hi(wr[4], wr[6], half); wb[3].y = comb_hi(wr[5], wr[7], half);

        // ---- GEMM2: embed^T(dims x rows) += C_chunk^T * W, A operand from LDS ----
        // Per K-group: stage all 8 operand pairs (2addr-combinable, halves 512B
        // apart), then 8 WMMAs across independent acc[dt] chains.
        #pragma unroll
        for (int j2 = 0; j2 < 4; ++j2) {
            const float* cA = cs + (4 * j2 + 2 * half) * DIMS + lm;
            v2f a[8];
            #pragma unroll
            for (int dt = 0; dt < 8; ++dt) {
                a[dt].x = cA[dt * 16];
                a[dt].y = cA[DIMS + dt * 16];
            }
            #pragma unroll
            for (int dt = 0; dt < 8; ++dt)
                acc[dt] = __builtin_amdgcn_wmma_f32_16x16x4_f32(
                              false, a[dt], false, wb[j2], (short)0, acc[dt], false, false);
        }

        __syncthreads();     // everyone done reading cstage[cur] before it is re-staged
    }

    // ---- finalize: normalize, combine halves, store ----
    const float wtot = wsum + shflx16(wsum);
    const float inv  = 1.f / wtot;
    {   // argmin cross-half combine (tie -> smaller index, matches first-occurrence)
        float od = shflx16(bestd);
        int   oi = __shfl_xor(besti, 16, 32);
        if (od < bestd || (od == bestd && oi < besti)) { bestd = od; besti = oi; }
    }
    // acc[dt][r] at lane L == embed[row lm][dim dt*16 + r + 8*half]
    float* orow = out_embed + (size_t)(m0 + lm) * DIMS + 8 * half;
    #pragma unroll
    for (int dt = 0; dt < 8; ++dt) {
        #pragma unroll
        for (int r = 0; r < 8; ++r)
            orow[dt * 16 + r] = acc[dt][r] * inv;
    }
    if (lane < 16) out_idx[m0 + lm] = (float)besti;
}

extern "C" void kernel_launch(void* const* d_in, const int* in_sizes, int n_in,
                              void* d_out, int out_size, void* d_ws, size_t ws_size,
                              hipStream_t stream) {
    const float* x  = (const float*)d_in[0];
    const float* cb = (const float*)d_in[1];
    float* out_embed = (float*)d_out;
    float* out_idx   = out_embed + (size_t)NROW * DIMS;
    const int blocks = NROW / (16 * WAVES_PER_BLK);  // 256 blocks x 4 waves = 1024 row-tiles
    softvq_kernel<<<blocks, 32 * WAVES_PER_BLK, 0, stream>>>(x, cb, out_embed, out_idx);
}